// PolicyGNN_2_80874234184327
// MI455X (gfx1250) — compile-verified
//
#include <hip/hip_runtime.h>
#include <hip/hip_bf16.h>

typedef __attribute__((ext_vector_type(2))) float v2f;
typedef __attribute__((ext_vector_type(8))) float v8f;

// ---------------------------------------------------------------------------
// Degree / normalization kernels
// ---------------------------------------------------------------------------
__global__ void k_deg_init(float* __restrict__ deg, int n) {
    int i = blockIdx.x * blockDim.x + threadIdx.x;
    if (i < n) deg[i] = 1.0f;  // self loop contributes 1
}

__global__ void k_deg_acc(const int* __restrict__ dst, float* __restrict__ deg, int e) {
    int i = blockIdx.x * blockDim.x + threadIdx.x;
    if (i < e) unsafeAtomicAdd(&deg[dst[i]], 1.0f);
}

__global__ void k_dinv(const float* __restrict__ deg, float* __restrict__ dinv, int n) {
    int i = blockIdx.x * blockDim.x + threadIdx.x;
    if (i < n) dinv[i] = rsqrtf(deg[i]);  // deg >= 1 always (self loops)
}

// ---------------------------------------------------------------------------
// Dense GEMM via V_WMMA_F32_16X16X4_F32.
// C[M,N] = A[M,K] @ B[K,N] (+ bias). One wave computes one 16x16 tile.
// Requires M % 16 == 0, K % 4 == 0, N % 16 == 0 (true here: M=50000=3125*16).
//
// A 16x4 f32 layout: lanes 0-15 / 16-31 both hold M=0..15;
//   vgpr0 = K = (2*half)+0, vgpr1 = K = (2*half)+1  (half = lane>>4)
// B 4x16 f32 layout (rows striped across lanes): lanes hold N=0..15;
//   vgpr0 = K row (2*half)+0, vgpr1 = K row (2*half)+1
// D 16x16 f32 layout: vgpr v -> M = v + 8*half, N = lane&15
// ---------------------------------------------------------------------------
__global__ __launch_bounds__(256) void k_gemm_wmma(
    const float* __restrict__ A, const float* __restrict__ B,
    const float* __restrict__ bias, float* __restrict__ C,
    int M, int K, int N) {
    int wave  = (int)((blockIdx.x * blockDim.x + threadIdx.x) >> 5);
    int lane  = threadIdx.x & 31;
    int tiles_n = N >> 4;
    int tiles_m = M >> 4;
    if (wave >= tiles_m * tiles_n) return;  // wave-uniform exit: EXEC stays all-1s
    int tm   = wave / tiles_n;
    int tn   = wave - tm * tiles_n;
    int row  = lane & 15;   // M index for A, N index for B/D
    int half = lane >> 4;

    const float* Arow = A + (size_t)(tm * 16 + row) * K + 2 * half;
    const float* Bcol = B + tn * 16 + row;

    v8f acc = {};
    for (int k0 = 0; k0 < K; k0 += 4) {
        v2f a = *(const v2f*)(Arow + k0);             // contiguous, 8B aligned
        v2f b;
        b.x = Bcol[(size_t)(k0 + 2 * half + 0) * N];
        b.y = Bcol[(size_t)(k0 + 2 * half + 1) * N];
        acc = __builtin_amdgcn_wmma_f32_16x16x4_f32(
            /*neg_a=*/false, a, /*neg_b=*/false, b,
            /*c_mod=*/(short)0, acc, /*reuse_a=*/false, /*reuse_b=*/false);
    }

    int col  = tn * 16 + row;
    float bv = bias ? bias[col] : 0.0f;
    float* Cout = C + (size_t)(tm * 16 + 8 * half) * N + col;
#pragma unroll
    for (int v = 0; v < 8; ++v) {
        Cout[(size_t)v * N] = acc[v] + bv;
    }
}

// ---------------------------------------------------------------------------
// Message passing (hidden dim fixed at 32, float4-vectorized, 8 thr/node-row)
// ---------------------------------------------------------------------------
__global__ void k_self_init(const float* __restrict__ h, const float* __restrict__ dinv,
                            float* __restrict__ out, int n) {
    int tid  = blockIdx.x * blockDim.x + threadIdx.x;
    int node = tid >> 3;
    if (node >= n) return;
    int j   = (tid & 7) * 4;
    float w = dinv[node];
    w *= w;  // self-loop norm = dinv^2
    float4 hv = *(const float4*)(h + (size_t)node * 32 + j);
    float4 ov = make_float4(hv.x * w, hv.y * w, hv.z * w, hv.w * w);
    *(float4*)(out + (size_t)node * 32 + j) = ov;
}

__global__ void k_edge_scatter(const float* __restrict__ h,
                               const int* __restrict__ src, const int* __restrict__ dst,
                               const float* __restrict__ dinv,
                               float* __restrict__ out, int e) {
    int tid = blockIdx.x * blockDim.x + threadIdx.x;
    int edge = tid >> 3;
    if (edge >= e) return;
    int j = (tid & 7) * 4;
    int s = src[edge];
    int d = dst[edge];
    float w = dinv[s] * dinv[d];
    float4 hv = *(const float4*)(h + (size_t)s * 32 + j);
    float* o = out + (size_t)d * 32 + j;
    unsafeAtomicAdd(o + 0, hv.x * w);
    unsafeAtomicAdd(o + 1, hv.y * w);
    unsafeAtomicAdd(o + 2, hv.z * w);
    unsafeAtomicAdd(o + 3, hv.w * w);
}

__global__ void k_bias_relu(const float* __restrict__ in, const float* __restrict__ b,
                            float* __restrict__ out, int total) {
    int i = blockIdx.x * blockDim.x + threadIdx.x;
    if (i >= total) return;
    float v = in[i] + b[i & 31];  // hidden dim = 32
    out[i] = fmaxf(v, 0.0f);
}

// ---------------------------------------------------------------------------
// Host launch
// ---------------------------------------------------------------------------
static inline int cdiv(long long a, long long b) { return (int)((a + b - 1) / b); }

extern "C" void kernel_launch(void* const* d_in, const int* in_sizes, int n_in,
                              void* d_out, int out_size, void* d_ws, size_t ws_size,
                              hipStream_t stream) {
    const float* x    = (const float*)d_in[0];
    const int*   ei   = (const int*)d_in[1];
    const float* W1   = (const float*)d_in[2];
    const float* b1   = (const float*)d_in[3];
    const float* W2   = (const float*)d_in[4];
    const float* b2   = (const float*)d_in[5];
    const float* Wout = (const float*)d_in[6];
    const float* bout = (const float*)d_in[7];
    float* out = (float*)d_out;

    const int H   = in_sizes[3];        // hidden = 32
    const int AO  = in_sizes[7];        // n_actions = 64
    const int FIN = in_sizes[2] / H;    // n_features = 128
    const int NN  = in_sizes[0] / FIN;  // 50000 nodes
    const int E   = in_sizes[1] / 2;    // 1.6M edges

    // workspace layout (floats): deg[NN] | dinv[NN] | bufA[NN*H] | bufB[NN*H]
    float* deg  = (float*)d_ws;
    float* dinv = deg + NN;
    float* bufA = dinv + NN;
    float* bufB = bufA + (size_t)NN * H;

    const int* srcE = ei;       // edge_index[0]
    const int* dstE = ei + E;   // edge_index[1]

    const int B256 = 256;

    // normalization (shared by both GCN layers)
    k_deg_init<<<cdiv(NN, B256), B256, 0, stream>>>(deg, NN);
    k_deg_acc<<<cdiv(E, B256), B256, 0, stream>>>(dstE, deg, E);
    k_dinv<<<cdiv(NN, B256), B256, 0, stream>>>(deg, dinv, NN);

    // ---- layer 1: h1 = relu(Aggregate(x @ W1) + b1) ----
    {
        long long waves = (long long)(NN >> 4) * (H >> 4);
        k_gemm_wmma<<<cdiv(waves * 32, B256), B256, 0, stream>>>(x, W1, nullptr, bufA, NN, FIN, H);
    }
    k_self_init<<<cdiv((long long)NN * 8, B256), B256, 0, stream>>>(bufA, dinv, bufB, NN);
    k_edge_scatter<<<cdiv((long long)E * 8, B256), B256, 0, stream>>>(bufA, srcE, dstE, dinv, bufB, E);
    k_bias_relu<<<cdiv((long long)NN * H, B256), B256, 0, stream>>>(bufB, b1, bufA, NN * H);

    // ---- layer 2: h2 = relu(Aggregate(h1 @ W2) + b2) ----
    {
        long long waves = (long long)(NN >> 4) * (H >> 4);
        k_gemm_wmma<<<cdiv(waves * 32, B256), B256, 0, stream>>>(bufA, W2, nullptr, bufB, NN, H, H);
    }
    k_self_init<<<cdiv((long long)NN * 8, B256), B256, 0, stream>>>(bufB, dinv, bufA, NN);
    k_edge_scatter<<<cdiv((long long)E * 8, B256), B256, 0, stream>>>(bufB, srcE, dstE, dinv, bufA, E);
    k_bias_relu<<<cdiv((long long)NN * H, B256), B256, 0, stream>>>(bufA, b2, bufB, NN * H);

    // ---- output head: logits = h2 @ Wout + bout ----
    {
        long long waves = (long long)(NN >> 4) * (AO >> 4);
        k_gemm_wmma<<<cdiv(waves * 32, B256), B256, 0, stream>>>(bufB, Wout, bout, out, NN, H, AO);
    }
}